// Encoder_60421599920740
// MI455X (gfx1250) — compile-verified
//
#include <hip/hip_runtime.h>
#include <hip/hip_bf16.h>

typedef __attribute__((ext_vector_type(2))) float v2f;
typedef __attribute__((ext_vector_type(8))) float v8f;

#define LDS_STRIDE 132  // 128 + 4 pad: bank = (4*row + k) % 64 -> conflict-free

// ---------------------------------------------------------------------------
// Zero-fill (float4 granularity; all our regions are multiples of 4 floats)
// ---------------------------------------------------------------------------
__global__ __launch_bounds__(256) void fill0_f4(float4* __restrict__ p, long n4) {
  long i = (long)blockIdx.x * blockDim.x + threadIdx.x;
  long stride = (long)gridDim.x * blockDim.x;
  float4 z = make_float4(0.f, 0.f, 0.f, 0.f);
  for (; i < n4; i += stride) p[i] = z;
}

// ---------------------------------------------------------------------------
// In-degree count (atomic f32 adds into L2-resident deg array)
// ---------------------------------------------------------------------------
__global__ __launch_bounds__(256) void degree_count(const int* __restrict__ dst,
                                                    float* __restrict__ deg, int nE) {
  int i = blockIdx.x * blockDim.x + threadIdx.x;
  int stride = gridDim.x * blockDim.x;
  for (; i < nE; i += stride) atomicAdd(&deg[dst[i]], 1.0f);
}

// deg -> 1/max(deg,1)  (in place)
__global__ __launch_bounds__(256) void inv_deg_kernel(float* __restrict__ deg, int n) {
  int i = blockIdx.x * blockDim.x + threadIdx.x;
  if (i < n) {
    float d = deg[i];
    deg[i] = 1.0f / fmaxf(d, 1.0f);
  }
}

// ---------------------------------------------------------------------------
// Edge scatter-accumulate: one wave32 per edge, each lane moves float4
// (128 feats = 32 lanes x 4). rowLimit lets layer-3 skip edges whose dst
// is outside the final batch.
// ---------------------------------------------------------------------------
__global__ __launch_bounds__(256) void scatter_accum(const float* __restrict__ h,
                                                     const int* __restrict__ src,
                                                     const int* __restrict__ dst,
                                                     float* __restrict__ agg,
                                                     int nE, int rowLimit) {
  const int lane = threadIdx.x & 31;
  int wave = (blockIdx.x * blockDim.x + threadIdx.x) >> 5;
  const int nWaves = (gridDim.x * blockDim.x) >> 5;
  for (int e = wave; e < nE; e += nWaves) {
    const int d = dst[e];  // wave-uniform
    if (d >= rowLimit) continue;
    const int s = src[e];
    const float4 v = *(const float4*)(h + (size_t)s * 128 + (size_t)lane * 4);
    float* p = agg + (size_t)d * 128 + (size_t)lane * 4;
    atomicAdd(p + 0, v.x);
    atomicAdd(p + 1, v.y);
    atomicAdd(p + 2, v.z);
    atomicAdd(p + 3, v.w);
  }
}

// ---------------------------------------------------------------------------
// Fused SAGE layer GEMM + bias + PReLU, full f32 via V_WMMA_F32_16X16X4_F32.
//   out[16 rows x 128 cols per block] =
//     prelu( (agg*invdeg) @ Wl + bl + h @ Wr )
// The 16-row A-stripes (mean-scaled agg, and h) are staged once per block
// into padded LDS and shared by all 8 waves (kills the 8x redundant global
// reads); invdeg is folded in during staging. Weights stream from global
// (each element read exactly once per block -> cache-friendly).
// A-frag (16x4 f32): lane m = lane&15; VGPR0/1 hold K = (lane>>4)*2 + {0,1}.
// B-frag (4x16 f32): lane n = lane&15; VGPR0/1 hold K = (lane>>4)*2 + {0,1}.
// C/D (16x16 f32): VGPR r -> row m0 + r + (lane>=16 ? 8 : 0), col = lane&15.
// ---------------------------------------------------------------------------
__global__ __launch_bounds__(256) void sage_wmma_gemm(const float* __restrict__ agg,
                                                      const float* __restrict__ invdeg,
                                                      const float* __restrict__ h,
                                                      const float* __restrict__ Wl,
                                                      const float* __restrict__ bvec,
                                                      const float* __restrict__ Wr,
                                                      const float* __restrict__ prelu_a,
                                                      float* __restrict__ out) {
  __shared__ float sMean[16 * LDS_STRIDE];
  __shared__ float sH[16 * LDS_STRIDE];

  const int lane = threadIdx.x & 31;
  const int wave = threadIdx.x >> 5;  // 0..7 -> 16-col block
  const int m0   = blockIdx.x << 4;   // 16 rows per block
  const int kh   = (lane >> 4) << 1;  // 0 or 2 (K sub-offset)
  const int col  = (wave << 4) + (lane & 15);

  // --- cooperative staging: 2 x 16x128 f32 tiles, float4 granularity -------
  // 512 float4 per tile, 256 threads -> 2 per thread per tile.
  for (int i = threadIdx.x; i < 512; i += 256) {
    const int r  = i >> 5;         // row within stripe
    const int c4 = (i & 31) << 2;  // col (float index, multiple of 4)
    const float idg = invdeg[m0 + r];
    float4 va = *(const float4*)(agg + (size_t)(m0 + r) * 128 + c4);
    float4 vh = *(const float4*)(h + (size_t)(m0 + r) * 128 + c4);
    va.x *= idg; va.y *= idg; va.z *= idg; va.w *= idg;  // fold mean divide
    *(float4*)(&sMean[r * LDS_STRIDE + c4]) = va;
    *(float4*)(&sH[r * LDS_STRIDE + c4])    = vh;
  }
  __syncthreads();

  const float* mRow = &sMean[(lane & 15) * LDS_STRIDE];
  const float* hRow = &sH[(lane & 15) * LDS_STRIDE];

  v8f acc = {};

#pragma unroll 8
  for (int k = 0; k < 128; k += 4) {
    const int kb = k + kh;
    // A fragments from LDS (ds_load_b64, conflict-free via padded stride)
    v2f am = *(const v2f*)(mRow + kb);
    v2f ah = *(const v2f*)(hRow + kb);
    // B fragments (row-major weights, stride 128 between K rows)
    v2f fwl, fwr;
    fwl.x = Wl[(size_t)kb * 128 + col];
    fwl.y = Wl[(size_t)(kb + 1) * 128 + col];
    fwr.x = Wr[(size_t)kb * 128 + col];
    fwr.y = Wr[(size_t)(kb + 1) * 128 + col];
    // D = A*B + C  (f32, 16x16x4)
    acc = __builtin_amdgcn_wmma_f32_16x16x4_f32(false, am, false, fwl,
                                                (short)0, acc, false, false);
    acc = __builtin_amdgcn_wmma_f32_16x16x4_f32(false, ah, false, fwr,
                                                (short)0, acc, false, false);
  }

  const float bias = bvec[col];
  const float a = prelu_a[0];
  const int rbase = m0 + ((lane >> 4) << 3);  // +8 row offset for upper half-wave
#pragma unroll
  for (int r = 0; r < 8; ++r) {
    float v = acc[r] + bias;
    v = (v >= 0.0f) ? v : a * v;
    out[(size_t)(rbase + r) * 128 + col] = v;
  }
}

// ---------------------------------------------------------------------------
// Orchestration
// ---------------------------------------------------------------------------
extern "C" void kernel_launch(void* const* d_in, const int* in_sizes, int n_in,
                              void* d_out, int out_size, void* d_ws, size_t ws_size,
                              hipStream_t stream) {
  (void)n_in; (void)ws_size;

  const float* x   = (const float*)d_in[0];
  const int*   ei  = (const int*)d_in[1];
  const float* W1l = (const float*)d_in[4];
  const float* b1l = (const float*)d_in[5];
  const float* W1r = (const float*)d_in[6];
  const float* W2l = (const float*)d_in[7];
  const float* b2l = (const float*)d_in[8];
  const float* W2r = (const float*)d_in[9];
  const float* W3l = (const float*)d_in[10];
  const float* b3l = (const float*)d_in[11];
  const float* W3r = (const float*)d_in[12];
  const float* pa  = (const float*)d_in[13];

  const int N = in_sizes[0] / 128;  // 100000
  const int E = in_sizes[1] / 2;    // 1600000
  const int B = out_size / 128;     // batch_size (4096), host-derivable
  const int* src = ei;              // edge_index row 0
  const int* dst = ei + E;          // edge_index row 1

  float* ws     = (float*)d_ws;
  float* invdeg = ws;                     // N floats
  float* agg    = invdeg + N;             // N*128
  float* hA     = agg + (size_t)N * 128;  // N*128
  float* hB     = hA + (size_t)N * 128;   // N*128

  const int T = 256;

  // in-degree -> 1/max(deg,1)
  fill0_f4<<<512, T, 0, stream>>>((float4*)invdeg, (long)N / 4);
  degree_count<<<2048, T, 0, stream>>>(dst, invdeg, E);
  inv_deg_kernel<<<(N + T - 1) / T, T, 0, stream>>>(invdeg, N);

  // layer 1: x -> hA
  fill0_f4<<<4096, T, 0, stream>>>((float4*)agg, (long)N * 32);
  scatter_accum<<<4096, T, 0, stream>>>(x, src, dst, agg, E, N);
  sage_wmma_gemm<<<N / 16, T, 0, stream>>>(agg, invdeg, x, W1l, b1l, W1r, pa, hA);

  // layer 2: hA -> hB
  fill0_f4<<<4096, T, 0, stream>>>((float4*)agg, (long)N * 32);
  scatter_accum<<<4096, T, 0, stream>>>(hA, src, dst, agg, E, N);
  sage_wmma_gemm<<<N / 16, T, 0, stream>>>(agg, invdeg, hA, W2l, b2l, W2r, pa, hB);

  // layer 3: only rows < B matter for the output
  fill0_f4<<<1024, T, 0, stream>>>((float4*)agg, (long)B * 32);
  scatter_accum<<<4096, T, 0, stream>>>(hB, src, dst, agg, E, B);
  sage_wmma_gemm<<<B / 16, T, 0, stream>>>(agg, invdeg, hB, W3l, b3l, W3r, pa,
                                           (float*)d_out);
}